// FGN_layer_47622597378049
// MI455X (gfx1250) — compile-verified
//
#include <hip/hip_runtime.h>

typedef float v2f __attribute__((ext_vector_type(2)));
typedef float v8f __attribute__((ext_vector_type(8)));

#define IN_F  1024
#define OUT_F 4096
#define BATCH 1024

#define MBLK 64
#define NBLK 128
#define KBLK 32
#define LSTR 36   // padded LDS row stride (floats): 16B-aligned staging, conflict-free b64 frag reads

// ---------------------------------------------------------------------------
// Prep: per output row i, bias[i] = -sum_j W_ij*C_ij, dvec[i] = sum_j (C_ij*ic_ij)^2
// ---------------------------------------------------------------------------
__global__ __launch_bounds__(256)
void fgn_prep(const float* __restrict__ W, const float* __restrict__ C,
              const float* __restrict__ IC,
              float* __restrict__ bias, float* __restrict__ dvec) {
    __shared__ float sb[256];
    __shared__ float sd[256];
    const int i = blockIdx.x;
    const int t = threadIdx.x;
    const float* w  = W  + (size_t)i * IN_F;
    const float* c  = C  + (size_t)i * IN_F;
    const float* ic = IC + (size_t)i * IN_F;
    float ab = 0.f, ad = 0.f;
    for (int j = t; j < IN_F; j += 256) {
        float wv = w[j], cv = c[j], iv = ic[j];
        ab += wv * cv;
        float u = cv * iv;
        ad += u * u;
    }
    sb[t] = ab; sd[t] = ad;
    __syncthreads();
    for (int s = 128; s > 0; s >>= 1) {
        if (t < s) { sb[t] += sb[t + s]; sd[t] += sd[t + s]; }
        __syncthreads();
    }
    if (t == 0) { bias[i] = -sb[0]; dvec[i] = sd[0]; }
}

// ---------------------------------------------------------------------------
// Epilogue helper: lane holds N = n (fixed), VGPR v holds M = m_base + v
// out = (acc_l + bias) * exp(-(acc_g + d))
// ---------------------------------------------------------------------------
__device__ __forceinline__
void fgn_epi(v8f al, v8f ag, int m_base, int n, float bn, float dn,
             float* __restrict__ out) {
#pragma unroll
    for (int v = 0; v < 8; ++v) {
        float l = al[v] + bn;
        float g = ag[v] + dn;
        out[(size_t)(m_base + v) * OUT_F + n] = l * __expf(-g);
    }
}

// ---------------------------------------------------------------------------
// Main fused kernel: block tile 64(M) x 128(N), 8 waves of 32, wave tile 32x32.
// Two f32 WMMA accumulators per 16x16 tile:
//   acc_l += x  (x) W^T
//   acc_g += x^2(x) ic2^T  +  x (x) (-2*c*ic2)^T
// ---------------------------------------------------------------------------
__global__ __launch_bounds__(256)
void fgn_main(const float* __restrict__ X,  const float* __restrict__ W,
              const float* __restrict__ C,  const float* __restrict__ IC,
              const float* __restrict__ bias, const float* __restrict__ dvec,
              float* __restrict__ out) {
    __shared__ float Xs[MBLK][LSTR];   // x tile          [m][k]
    __shared__ float Ws[NBLK][LSTR];   // W tile          [n][k]
    __shared__ float Gs[NBLK][LSTR];   // ic^2 tile       [n][k]
    __shared__ float Cs[NBLK][LSTR];   // -2*c*ic^2 tile  [n][k]

    const int t    = threadIdx.x;
    const int lane = t & 31;
    const int r    = lane & 15;   // row-within-16 for A/B frags, N index for C/D
    const int h    = lane >> 4;   // half select: K pair (frags) / M+8 (C,D)
    const int wv   = t >> 5;      // wave 0..7
    const int wm   = (wv & 1) * 32;   // wave M offset in block tile
    const int wn   = (wv >> 1) * 32;  // wave N offset in block tile

    const int nb = blockIdx.x * NBLK;
    const int mb = blockIdx.y * MBLK;

    v8f accl00 = {}, accl01 = {}, accl10 = {}, accl11 = {};
    v8f accg00 = {}, accg01 = {}, accg10 = {}, accg11 = {};

    const int cg = (t & 7) * 4;   // k-column granule within K-block (float4)
    const int rr = t >> 3;        // 0..31 staging row

    for (int kb = 0; kb < IN_F; kb += KBLK) {
        // ---- stage X (64 x 32) ----
#pragma unroll
        for (int i = 0; i < MBLK / 32; ++i) {
            const int row = rr + 32 * i;
            float4 v = *(const float4*)&X[(size_t)(mb + row) * IN_F + kb + cg];
            *(float4*)&Xs[row][cg] = v;
        }
        // ---- stage W, derive ic^2 and -2*c*ic^2 (128 x 32 each) ----
#pragma unroll
        for (int i = 0; i < NBLK / 32; ++i) {
            const int row = rr + 32 * i;
            const size_t off = (size_t)(nb + row) * IN_F + kb + cg;
            float4 w4 = *(const float4*)&W[off];
            float4 c4 = *(const float4*)&C[off];
            float4 i4 = *(const float4*)&IC[off];
            float4 g4, q4;
            g4.x = i4.x * i4.x; g4.y = i4.y * i4.y;
            g4.z = i4.z * i4.z; g4.w = i4.w * i4.w;
            q4.x = -2.f * c4.x * g4.x; q4.y = -2.f * c4.y * g4.y;
            q4.z = -2.f * c4.z * g4.z; q4.w = -2.f * c4.w * g4.w;
            *(float4*)&Ws[row][cg] = w4;
            *(float4*)&Gs[row][cg] = g4;
            *(float4*)&Cs[row][cg] = q4;
        }
        __syncthreads();

        // ---- 8 x (K=4) WMMA steps over this K-block ----
#pragma unroll
        for (int kk = 0; kk < KBLK; kk += 4) {
            const int kc = kk + 2 * h;
            v2f a0 = *(const v2f*)&Xs[wm + r][kc];
            v2f a1 = *(const v2f*)&Xs[wm + 16 + r][kc];
            v2f s0 = a0 * a0;
            v2f s1 = a1 * a1;
            v2f bw0 = *(const v2f*)&Ws[wn + r][kc];
            v2f bw1 = *(const v2f*)&Ws[wn + 16 + r][kc];
            v2f bg0 = *(const v2f*)&Gs[wn + r][kc];
            v2f bg1 = *(const v2f*)&Gs[wn + 16 + r][kc];
            v2f bc0 = *(const v2f*)&Cs[wn + r][kc];
            v2f bc1 = *(const v2f*)&Cs[wn + 16 + r][kc];

            // linear part: x (x) W^T
            accl00 = __builtin_amdgcn_wmma_f32_16x16x4_f32(false, a0, false, bw0, (short)0, accl00, false, false);
            accl01 = __builtin_amdgcn_wmma_f32_16x16x4_f32(false, a0, false, bw1, (short)0, accl01, false, false);
            accl10 = __builtin_amdgcn_wmma_f32_16x16x4_f32(false, a1, false, bw0, (short)0, accl10, false, false);
            accl11 = __builtin_amdgcn_wmma_f32_16x16x4_f32(false, a1, false, bw1, (short)0, accl11, false, false);
            // gaussian part: x^2 (x) ic2^T
            accg00 = __builtin_amdgcn_wmma_f32_16x16x4_f32(false, s0, false, bg0, (short)0, accg00, false, false);
            accg01 = __builtin_amdgcn_wmma_f32_16x16x4_f32(false, s0, false, bg1, (short)0, accg01, false, false);
            accg10 = __builtin_amdgcn_wmma_f32_16x16x4_f32(false, s1, false, bg0, (short)0, accg10, false, false);
            accg11 = __builtin_amdgcn_wmma_f32_16x16x4_f32(false, s1, false, bg1, (short)0, accg11, false, false);
            // gaussian part: x (x) (-2 c ic2)^T
            accg00 = __builtin_amdgcn_wmma_f32_16x16x4_f32(false, a0, false, bc0, (short)0, accg00, false, false);
            accg01 = __builtin_amdgcn_wmma_f32_16x16x4_f32(false, a0, false, bc1, (short)0, accg01, false, false);
            accg10 = __builtin_amdgcn_wmma_f32_16x16x4_f32(false, a1, false, bc0, (short)0, accg10, false, false);
            accg11 = __builtin_amdgcn_wmma_f32_16x16x4_f32(false, a1, false, bc1, (short)0, accg11, false, false);
        }
        __syncthreads();
    }

    // ---- epilogue: out = (l + bias) * exp(-(g + d)) ----
    const int n0  = nb + wn + r;
    const int m0e = mb + wm + 8 * h;
    const float bn0 = bias[n0],      dn0 = dvec[n0];
    const float bn1 = bias[n0 + 16], dn1 = dvec[n0 + 16];
    fgn_epi(accl00, accg00, m0e,      n0,      bn0, dn0, out);
    fgn_epi(accl01, accg01, m0e,      n0 + 16, bn1, dn1, out);
    fgn_epi(accl10, accg10, m0e + 16, n0,      bn0, dn0, out);
    fgn_epi(accl11, accg11, m0e + 16, n0 + 16, bn1, dn1, out);
}

// ---------------------------------------------------------------------------
extern "C" void kernel_launch(void* const* d_in, const int* in_sizes, int n_in,
                              void* d_out, int out_size, void* d_ws, size_t ws_size,
                              hipStream_t stream) {
    const float* x   = (const float*)d_in[0];
    const float* wgt = (const float*)d_in[1];
    const float* cen = (const float*)d_in[2];
    const float* ic  = (const float*)d_in[3];
    float* bias = (float*)d_ws;          // OUT_F floats
    float* dvec = bias + OUT_F;          // OUT_F floats
    float* out  = (float*)d_out;

    fgn_prep<<<OUT_F, 256, 0, stream>>>(wgt, cen, ic, bias, dvec);

    dim3 grid(OUT_F / NBLK, BATCH / MBLK);
    fgn_main<<<grid, 256, 0, stream>>>(x, wgt, cen, ic, bias, dvec, out);
}